// ProposalGaussian_43482248905252
// MI455X (gfx1250) — compile-verified
//
#include <hip/hip_runtime.h>
#include <math.h>

typedef __attribute__((ext_vector_type(2))) float v2f;
typedef __attribute__((ext_vector_type(4))) float v4f;
typedef __attribute__((ext_vector_type(8))) float v8f;

#define WMMA_F32(a, b, c) \
  __builtin_amdgcn_wmma_f32_16x16x4_f32(false, (a), false, (b), (short)0, (c), false, false)

__device__ __forceinline__ float sigmoidf_(float v) { return 1.0f / (1.0f + expf(-v)); }

// Per-wave LDS slice: H [32][17] + h1 [16][33] + raw [32][25] = 1872 floats
#define WAVE_LDS 1872

__global__ __launch_bounds__(256) void proposal_gaussian_kernel(
    const float* __restrict__ xin, const int* __restrict__ hashidxs,
    const float* __restrict__ table, const float* __restrict__ w1,
    const float* __restrict__ b1, const float* __restrict__ w2,
    const float* __restrict__ b2, float* __restrict__ out, int N)
{
  constexpr int LVLS = 8;
  constexpr unsigned TMASK = (1u << 14) - 1u;
  constexpr unsigned PRIME = 2654435761u;
  // floor(16 * (W/16)^(l/7)), floor(16 * (H/16)^(l/7)) precomputed (float64 host math)
  const int RX[LVLS] = {16, 28, 52, 95, 172, 312, 565, 1024};
  const int RY[LVLS] = {16, 27, 48, 84, 146, 254, 441, 768};

  __shared__ float smem[8 * WAVE_LDS];

  const int tid   = threadIdx.x;
  const int lane  = tid & 31;
  const int wid   = tid >> 5;
  const int lhalf = lane >> 4;   // 0: lanes 0-15, 1: lanes 16-31
  const int lcol  = lane & 15;

  float* Hl   = smem + wid * WAVE_LDS;  // [32][17]  H features (A-matrix source, layer 1)
  float* h1l  = Hl + 32 * 17;           // [16][33]  hidden acts (A-matrix source, layer 2)
  float* rawl = h1l + 16 * 33;          // [32][25]  raw MLP outputs, per-point rows

  const long long p  = (long long)blockIdx.x * 256 + (long long)(wid * 32 + lane);
  const bool valid   = p < (long long)N;
  const long long pc = valid ? p : (long long)(N - 1);

  float* out_mu = out;                       // mu        N x 8
  float* out_is = out + (size_t)N * 8;       // inv_sigma N x 8
  float* out_w  = out + (size_t)N * 16;      // weight    N x 8
  float* out_H  = out + (size_t)N * 24;      // H         N x 16

  // ---------------- preload weight B-fragments into registers ----------------
  // B 4x16 f32 layout: VGPR0 = rows {0,2} (lane halves), VGPR1 = rows {1,3}
  v2f bf1[4][2];
  #pragma unroll
  for (int ks = 0; ks < 4; ++ks) {
    int r = 4 * ks + 2 * lhalf;
    #pragma unroll
    for (int nt = 0; nt < 2; ++nt) {
      int c = nt * 16 + lcol;
      bf1[ks][nt].x = w1[r * 32 + c];
      bf1[ks][nt].y = w1[(r + 1) * 32 + c];
    }
  }
  v2f bf2[8][2];
  #pragma unroll
  for (int ks = 0; ks < 8; ++ks) {
    int r = 4 * ks + 2 * lhalf;
    #pragma unroll
    for (int nt = 0; nt < 2; ++nt) {
      int c = nt * 16 + lcol;
      bool ok = c < 24;                 // pad w2 columns 24..31 with zeros
      bf2[ks][nt].x = ok ? w2[r * 24 + c] : 0.0f;
      bf2[ks][nt].y = ok ? w2[(r + 1) * 24 + c] : 0.0f;
    }
  }
  const float b1n0 = b1[lcol];
  const float b1n1 = b1[16 + lcol];
  const float b2n0 = b2[lcol];
  const float b2n1 = (lcol < 8) ? b2[16 + lcol] : 0.0f;

  // ---------------- per-lane multiresolution hash encoding ----------------
  const float ptx = xin[2 * pc + 0];
  const float pty = xin[2 * pc + 1];
  const float2* tb = (const float2*)table + (size_t)hashidxs[pc] * ((size_t)LVLS << 14);

  float Hreg[16];
  #pragma unroll
  for (int l = 0; l < LVLS; ++l) {
    float px = ptx * (float)RX[l];
    float py = pty * (float)RY[l];
    float fx = floorf(px), fy = floorf(py);
    float wx = px - fx, wy = py - fy;
    unsigned cx = (unsigned)(int)fx;
    unsigned cy = (unsigned)(int)fy;
    const float2* tl = tb + ((size_t)l << 14);
    unsigned hy0 = cy * PRIME;
    unsigned hy1 = (cy + 1u) * PRIME;
    float2 f00 = tl[(cx        ^ hy0) & TMASK];   // L2-resident gathers (table = 20MB << 192MB L2)
    float2 f10 = tl[((cx + 1u) ^ hy0) & TMASK];
    float2 f01 = tl[(cx        ^ hy1) & TMASK];
    float2 f11 = tl[((cx + 1u) ^ hy1) & TMASK];
    float w00 = (1.0f - wx) * (1.0f - wy);
    float w10 = wx * (1.0f - wy);
    float w01 = (1.0f - wx) * wy;
    float w11 = wx * wy;
    Hreg[2 * l]     = f00.x * w00 + f10.x * w10 + f01.x * w01 + f11.x * w11;
    Hreg[2 * l + 1] = f00.y * w00 + f10.y * w10 + f01.y * w01 + f11.y * w11;
  }

  // stage H into LDS for WMMA; stream H output non-temporally (don't pollute L2)
  #pragma unroll
  for (int i = 0; i < 16; ++i) Hl[lane * 17 + i] = Hreg[i];
  if (valid) {
    v4f* ho = (v4f*)(out_H + (size_t)p * 16);
    #pragma unroll
    for (int q = 0; q < 4; ++q) {
      v4f v = {Hreg[4 * q], Hreg[4 * q + 1], Hreg[4 * q + 2], Hreg[4 * q + 3]};
      __builtin_nontemporal_store(v, ho + q);
    }
  }
  __builtin_amdgcn_wave_barrier();   // LDS in-order within wave; fence the scheduler

  // ---------------- MLP via f32 WMMA, two 16-point tiles per wave ----------------
  #pragma unroll
  for (int t = 0; t < 2; ++t) {
    // layer 1: (16x16) x (16x32) = 4 K-steps x 2 N-tiles
    v8f acc0 = {0.f,0.f,0.f,0.f,0.f,0.f,0.f,0.f};
    v8f acc1 = {0.f,0.f,0.f,0.f,0.f,0.f,0.f,0.f};
    #pragma unroll
    for (int ks = 0; ks < 4; ++ks) {
      int k = 4 * ks + 2 * lhalf;      // A 16x4: lanes 0-15 K{0,1}, lanes 16-31 K{2,3}
      v2f a;
      a.x = Hl[(t * 16 + lcol) * 17 + k];
      a.y = Hl[(t * 16 + lcol) * 17 + k + 1];
      acc0 = WMMA_F32(a, bf1[ks][0], acc0);
      acc1 = WMMA_F32(a, bf1[ks][1], acc1);
    }
    // gaussian activation exp(-50 x^2); restage as layer-2 A source
    #pragma unroll
    for (int v = 0; v < 8; ++v) {
      int m = 8 * lhalf + v;           // C/D: VGPR v holds row v (+8 for upper lanes)
      float z0 = acc0[v] + b1n0;
      float z1 = acc1[v] + b1n1;
      h1l[m * 33 + lcol]      = expf(-50.0f * z0 * z0);
      h1l[m * 33 + 16 + lcol] = expf(-50.0f * z1 * z1);
    }
    __builtin_amdgcn_wave_barrier();

    // layer 2: (16x32) x (32x24 padded to 32) = 8 K-steps x 2 N-tiles
    v8f r0 = {0.f,0.f,0.f,0.f,0.f,0.f,0.f,0.f};
    v8f r1 = {0.f,0.f,0.f,0.f,0.f,0.f,0.f,0.f};
    #pragma unroll
    for (int ks = 0; ks < 8; ++ks) {
      int k = 4 * ks + 2 * lhalf;
      v2f a;
      a.x = h1l[lcol * 33 + k];
      a.y = h1l[lcol * 33 + k + 1];
      r0 = WMMA_F32(a, bf2[ks][0], r0);
      r1 = WMMA_F32(a, bf2[ks][1], r1);
    }
    // bias + park raw rows per point
    #pragma unroll
    for (int v = 0; v < 8; ++v) {
      int m = t * 16 + 8 * lhalf + v;
      rawl[m * 25 + lcol] = r0[v] + b2n0;
      if (lcol < 8) rawl[m * 25 + 16 + lcol] = r1[v] + b2n1;
    }
    __builtin_amdgcn_wave_barrier();
  }

  // ---------------- per-point nonlinearities + coalesced NT stores ----------------
  if (valid) {
    float rv[24];
    #pragma unroll
    for (int i = 0; i < 24; ++i) rv[i] = rawl[lane * 25 + i];

    v4f mu0 = {sigmoidf_(rv[8]),  sigmoidf_(rv[9]),  sigmoidf_(rv[10]), sigmoidf_(rv[11])};
    v4f mu1 = {sigmoidf_(rv[12]), sigmoidf_(rv[13]), sigmoidf_(rv[14]), sigmoidf_(rv[15])};
    v4f is0 = {expf(rv[16]), expf(rv[17]), expf(rv[18]), expf(rv[19])};
    v4f is1 = {expf(rv[20]), expf(rv[21]), expf(rv[22]), expf(rv[23])};
    v4f wt0 = {expf(rv[0]),  expf(rv[1]),  expf(rv[2]),  expf(rv[3])};
    v4f wt1 = {expf(rv[4]),  expf(rv[5]),  expf(rv[6]),  expf(rv[7])};

    v4f* pm = (v4f*)(out_mu + (size_t)p * 8);
    v4f* pi = (v4f*)(out_is + (size_t)p * 8);
    v4f* pw = (v4f*)(out_w  + (size_t)p * 8);
    __builtin_nontemporal_store(mu0, pm);
    __builtin_nontemporal_store(mu1, pm + 1);
    __builtin_nontemporal_store(is0, pi);
    __builtin_nontemporal_store(is1, pi + 1);
    __builtin_nontemporal_store(wt0, pw);
    __builtin_nontemporal_store(wt1, pw + 1);
  }
}

extern "C" void kernel_launch(void* const* d_in, const int* in_sizes, int n_in,
                              void* d_out, int out_size, void* d_ws, size_t ws_size,
                              hipStream_t stream) {
  const float* x        = (const float*)d_in[0];
  const int*   hashidxs = (const int*)d_in[1];
  // d_in[2] = color (unused by the reference outputs)
  const float* table    = (const float*)d_in[3];
  const float* w1       = (const float*)d_in[4];
  const float* b1       = (const float*)d_in[5];
  const float* w2       = (const float*)d_in[6];
  const float* b2       = (const float*)d_in[7];

  const int N = in_sizes[1];                  // hashidxs count == number of points
  const int blocks = (N + 255) / 256;         // 256 threads = 8 wave32, 32 points/wave

  proposal_gaussian_kernel<<<blocks, 256, 0, stream>>>(
      x, hashidxs, table, w1, b1, w2, b2, (float*)d_out, N);
}